// PathConv_5059471475167
// MI455X (gfx1250) — compile-verified
//
#include <hip/hip_runtime.h>

// CDNA5 / gfx1250. Wave32. WMMA fp32 16x16x4.
typedef __attribute__((ext_vector_type(2))) float v2f;
typedef __attribute__((ext_vector_type(8))) float v8f;

#define WAVES_PER_BLOCK 4
#define E_STRIDE 84     // 16x80 edge tile; 84 mod 64 = 20 -> conflict-free A-frag reads
#define W1_PSTRIDE 160  // 40 row-pairs x (64 cols x 2); +160 = +32 banks -> hi half disjoint
#define W2_PSTRIDE 96   // 32 row-pairs x (32 cols x 2); +96  = +32 banks -> hi half disjoint
#define H_STRIDE 20     // 16x16 h slab

// order-preserving float->uint encoding for atomic max (non-NaN)
__device__ __forceinline__ unsigned fenc(float f) {
  unsigned u = __float_as_uint(f);
  return (u & 0x80000000u) ? ~u : (u | 0x80000000u);
}
__device__ __forceinline__ float fdec(unsigned u) {
  return (u & 0x80000000u) ? __uint_as_float(u & 0x7fffffffu) : __uint_as_float(~u);
}
__device__ __forceinline__ float lrelu(float v) { return fmaxf(v, 0.01f * v); }

__global__ void pathconv_init(unsigned* __restrict__ outu, long long n) {
  long long i = (long long)blockIdx.x * blockDim.x + threadIdx.x;
  if (i < n) outu[i] = 0u;  // sentinel: "empty segment"
}

__global__ void pathconv_final(unsigned* __restrict__ outu, const float* __restrict__ x,
                               long long n) {
  long long i = (long long)blockIdx.x * blockDim.x + threadIdx.x;
  if (i < n) {
    unsigned u = outu[i];
    float agg = (u == 0u) ? 0.0f : fdec(u);  // empty -> 0 (ref: isneginf -> 0)
    ((float*)outu)[i] = fmaxf(agg, x[i]);    // OUT_CH == IN_CH -> x slice == x
  }
}

__global__ void __launch_bounds__(WAVES_PER_BLOCK * 32)
pathconv_edges(const float* __restrict__ x, const long long* __restrict__ ei,
               const float* __restrict__ ea, const float* __restrict__ W1,
               const float* __restrict__ b1, const float* __restrict__ W2,
               const float* __restrict__ b2, unsigned* __restrict__ outu,
               long long nEdges) {
  // pair-interleaved weights: Wi[p][2c+q] = Wpad[2p+q][c]  -> B frag = one b64 load
  __shared__ __align__(16) float W1i[40 * W1_PSTRIDE];  // padded 80x64 (cols 56..63 = 0)
  __shared__ __align__(16) float W2i[32 * W2_PSTRIDE];  // padded 64x32 (rows 56..63 = 0)
  __shared__ __align__(16) float Es[WAVES_PER_BLOCK][16 * E_STRIDE];
  __shared__ __align__(16) float Hs[WAVES_PER_BLOCK][16 * H_STRIDE];
  __shared__ int colS[WAVES_PER_BLOCK][16];

  const int tid = threadIdx.x;
  const int lane = tid & 31;
  const int wslot = tid >> 5;
  const int m = lane & 15;  // A-row within 16-edge tile / B,C column within N tile
  const int g = lane >> 4;  // half-wave group
  const int n = m;

  // ---- stage pair-interleaved padded weights once per block ----
  for (int i = tid; i < 80 * 64; i += WAVES_PER_BLOCK * 32) {
    int r = i >> 6, c = i & 63;
    W1i[(r >> 1) * W1_PSTRIDE + c * 2 + (r & 1)] = (c < 56) ? W1[r * 56 + c] : 0.0f;
  }
  for (int i = tid; i < 64 * 32; i += WAVES_PER_BLOCK * 32) {
    int r = i >> 5, c = i & 31;
    W2i[(r >> 1) * W2_PSTRIDE + c * 2 + (r & 1)] = (r < 56) ? W2[r * 32 + c] : 0.0f;
  }
  __syncthreads();

  // lane-resident biases (depend only on n)
  float b1v[4], b2v[2];
#pragma unroll
  for (int nt = 0; nt < 4; ++nt) {
    int c = nt * 16 + n;
    b1v[nt] = (c < 56) ? b1[c] : 0.0f;
  }
#pragma unroll
  for (int nt = 0; nt < 2; ++nt) b2v[nt] = b2[nt * 16 + n];

  const long long nTiles = (nEdges + 15) >> 4;
  const long long waveId = (long long)blockIdx.x * WAVES_PER_BLOCK + wslot;
  const long long waveStride = (long long)gridDim.x * WAVES_PER_BLOCK;

  for (long long t = waveId; t < nTiles; t += waveStride) {
    __builtin_amdgcn_wave_barrier();  // prev tile's LDS reads before restaging
    // ---- stage 16x80 edge tile: [x[rol] | x[col] | edge_attr] ----
    const long long e = t * 16 + m;
    if (e < nEdges) {
      long long node = ei[(g ? nEdges : 0) + e];  // g=0: rol, g=1: col
      const float4* src = (const float4*)(x + node * 32);
#pragma unroll
      for (int j = 0; j < 8; ++j)
        *(float4*)&Es[wslot][m * E_STRIDE + g * 32 + 4 * j] = src[j];
      if (g) {
        const float4* eav = (const float4*)(ea + e * 16);
#pragma unroll
        for (int j = 0; j < 4; ++j)
          *(float4*)&Es[wslot][m * E_STRIDE + 64 + 4 * j] = eav[j];
        colS[wslot][m] = (int)node;
      }
    } else {
      float4 z4 = {0.f, 0.f, 0.f, 0.f};
#pragma unroll
      for (int j = 0; j < 8; ++j)
        *(float4*)&Es[wslot][m * E_STRIDE + g * 32 + 4 * j] = z4;
      if (g) {
#pragma unroll
        for (int j = 0; j < 4; ++j)
          *(float4*)&Es[wslot][m * E_STRIDE + 64 + 4 * j] = z4;
        colS[wslot][m] = -1;
      }
    }
    __builtin_amdgcn_wave_barrier();  // staging stores before fragment reads (LDS in-order in HW)

    // ---- layer 1: [16x80] x [80x64] via wmma f32 16x16x4 ----
    v8f zero = {0.f, 0.f, 0.f, 0.f, 0.f, 0.f, 0.f, 0.f};
    v8f acc1[4] = {zero, zero, zero, zero};
    for (int k = 0; k < 20; ++k) {
      // A frag: lanes0-15 K=4k,4k+1 ; lanes16-31 K=4k+2,4k+3 (even offset -> b64)
      v2f a = *(const v2f*)&Es[wslot][m * E_STRIDE + 4 * k + 2 * g];
      int p = 2 * k + g;  // row pair (4k+2g)/2
#pragma unroll
      for (int nt = 0; nt < 4; ++nt) {
        v2f b = *(const v2f*)&W1i[p * W1_PSTRIDE + (nt * 16 + n) * 2];
        acc1[nt] = __builtin_amdgcn_wmma_f32_16x16x4_f32(
            false, a, false, b, (short)0, acc1[nt], false, false);
      }
    }

    // ---- layer 2 fused: bounce each 16-wide h slice through LDS slab ----
    v8f acc2[2] = {zero, zero};
#pragma unroll
    for (int nt1 = 0; nt1 < 4; ++nt1) {
      float bb = b1v[nt1];
#pragma unroll
      for (int r = 0; r < 8; ++r) {
        float hv = lrelu(acc1[nt1][r] + bb);  // C frag (r, lane) = h[r+8g][16*nt1+n]
        Hs[wslot][(r + 8 * g) * H_STRIDE + n] = hv;
      }
      __builtin_amdgcn_wave_barrier();
#pragma unroll
      for (int k2 = 0; k2 < 4; ++k2) {
        v2f a2 = *(const v2f*)&Hs[wslot][m * H_STRIDE + 4 * k2 + 2 * g];
        int p2 = nt1 * 8 + 2 * k2 + g;  // row pair (nt1*16+4k2+2g)/2
#pragma unroll
        for (int nt2 = 0; nt2 < 2; ++nt2) {
          v2f b2 = *(const v2f*)&W2i[p2 * W2_PSTRIDE + (nt2 * 16 + n) * 2];
          acc2[nt2] = __builtin_amdgcn_wmma_f32_16x16x4_f32(
              false, a2, false, b2, (short)0, acc2[nt2], false, false);
        }
      }
      __builtin_amdgcn_wave_barrier();  // slab reads done before next slice overwrites
    }

    // ---- bias + lrelu + scatter-max (ordered-uint atomic max) ----
#pragma unroll
    for (int nt2 = 0; nt2 < 2; ++nt2) {
      float bb2 = b2v[nt2];
#pragma unroll
      for (int r = 0; r < 8; ++r) {
        float v = lrelu(acc2[nt2][r] + bb2);
        int dst = colS[wslot][r + 8 * g];
        if (dst >= 0)
          atomicMax(&outu[(long long)dst * 32 + nt2 * 16 + n], fenc(v));
      }
    }
  }
}

extern "C" void kernel_launch(void* const* d_in, const int* in_sizes, int n_in,
                              void* d_out, int out_size, void* d_ws, size_t ws_size,
                              hipStream_t stream) {
  const float* x = (const float*)d_in[0];
  const long long* ei = (const long long*)d_in[1];
  const float* ea = (const float*)d_in[2];
  const float* W1 = (const float*)d_in[3];
  const float* b1 = (const float*)d_in[4];
  const float* W2 = (const float*)d_in[5];
  const float* b2 = (const float*)d_in[6];
  unsigned* outu = (unsigned*)d_out;

  const long long nEdges = (long long)in_sizes[1] / 2;
  const long long nOut = (long long)out_size;

  int eltBlocks = (int)((nOut + 255) / 256);
  if (eltBlocks < 1) eltBlocks = 1;
  pathconv_init<<<eltBlocks, 256, 0, stream>>>(outu, nOut);

  long long nTiles = (nEdges + 15) / 16;
  long long blocks = (nTiles + WAVES_PER_BLOCK * 4 - 1) / (WAVES_PER_BLOCK * 4);
  if (blocks < 1) blocks = 1;
  if (blocks > 65535) blocks = 65535;
  pathconv_edges<<<(int)blocks, WAVES_PER_BLOCK * 32, 0, stream>>>(
      x, ei, ea, W1, b1, W2, b2, outu, nEdges);

  pathconv_final<<<eltBlocks, 256, 0, stream>>>(outu, x, nOut);
}